// HyperbolicPrototypeHead_64750926955055
// MI455X (gfx1250) — compile-verified
//
#include <hip/hip_runtime.h>
#include <math.h>

// Hyperbolic prototype head: B=2048, C=512, D=256.
// dist(b,c) depends only on ||z_b||^2, ||p_c||^2 and z_b.p_c  ->  f32 WMMA GEMM
// ([B,256] x [256,C]) with a transcendental epilogue.
// Kernel 2 uses 2x2 register tiling: 4 v_wmma_f32_16x16x4_f32 per K-step on
// 4 b64 loads (8 FLOP/byte from cache, 2x the reuse of a 1-tile wave).

typedef float v2f __attribute__((ext_vector_type(2)));
typedef float v8f __attribute__((ext_vector_type(8)));

__device__ __forceinline__ float current_c(float raw) {
  // softplus(raw) + 1e-6, clipped to [1e-4, 1e4]; numerically-stable softplus.
  float sp = fmaxf(raw, 0.0f) + log1pf(expf(-fabsf(raw)));
  float c = sp + 1e-6f;
  return fminf(fmaxf(c, 1e-4f), 1e4f);
}

// ---------------------------------------------------------------------------
// Kernel 1: expmap0 for both matrices. One wave (32 lanes) per row of 256 f32.
// Writes scaled rows Z/P and squared ball-norms X2/Y2 (= tanh^2(sc*||u||)/c).
// ---------------------------------------------------------------------------
__global__ __launch_bounds__(256) void expmap0_prep(
    const float* __restrict__ feats,   // [B, 256]
    const float* __restrict__ proto,   // [C, 256]
    const float* __restrict__ raw_c,   // [1]
    float* __restrict__ Z,             // [B, 256]
    float* __restrict__ P,             // [C, 256]
    float* __restrict__ X2,            // [B]
    float* __restrict__ Y2,            // [C]
    int B, int C) {
  const int D = 256;
  int wave = (blockIdx.x * blockDim.x + threadIdx.x) >> 5;
  int lane = threadIdx.x & 31;
  if (wave >= B + C) return;

  const float* src;
  float* dst;
  float* n2out;
  if (wave < B) {
    src = feats + (size_t)wave * D;
    dst = Z + (size_t)wave * D;
    n2out = X2 + wave;
  } else {
    int r = wave - B;
    src = proto + (size_t)r * D;
    dst = P + (size_t)r * D;
    n2out = Y2 + r;
  }

  // 8 contiguous floats per lane (256 / 32).
  float4 a = *(const float4*)(src + lane * 8);
  float4 b = *(const float4*)(src + lane * 8 + 4);
  float s = a.x * a.x + a.y * a.y + a.z * a.z + a.w * a.w +
            b.x * b.x + b.y * b.y + b.z * b.z + b.w * b.w;
  // wave32 butterfly reduction
  #pragma unroll
  for (int off = 16; off > 0; off >>= 1) s += __shfl_xor(s, off, 32);

  float c = current_c(raw_c[0]);
  float sc = sqrtf(c);
  float norm = fmaxf(sqrtf(s), 1e-15f);
  float th = tanhf(sc * norm);
  float scale = th / (sc * norm);

  float4 oa, ob;
  oa.x = a.x * scale; oa.y = a.y * scale; oa.z = a.z * scale; oa.w = a.w * scale;
  ob.x = b.x * scale; ob.y = b.y * scale; ob.z = b.z * scale; ob.w = b.w * scale;
  *(float4*)(dst + lane * 8) = oa;
  *(float4*)(dst + lane * 8 + 4) = ob;
  if (lane == 0) *n2out = (th * th) / c;  // || expmap0(u) ||^2
}

// ---------------------------------------------------------------------------
// Kernel 2: WMMA f32 GEMM + hyperbolic epilogue, 32x32 tile per wave.
// A-matrix (16x4 f32): lane l holds (M=l%16, K=k+2*(l/16) .. +1) -> b64 load.
// B-matrix (4x16 f32): lane l holds (N=l%16, K=k+2*(l/16) .. +1) -> b64 load.
// Block = 4 waves (2x2), covering a 64x64 macro-tile.
// ---------------------------------------------------------------------------
__global__ __launch_bounds__(128) void hyper_gemm_dist(
    const float* __restrict__ Z,   // [B, 256]
    const float* __restrict__ P,   // [C, 256]
    const float* __restrict__ X2,  // [B]
    const float* __restrict__ Y2,  // [C]
    const float* __restrict__ raw_c,
    float* __restrict__ out,       // [B, C]
    int Cn) {
  const int D = 256;
  int lane = threadIdx.x;           // 0..31
  int wave = threadIdx.y;           // 0..3
  int l16 = lane & 15;
  int hi = lane >> 4;               // 0 for lanes 0-15, 1 for lanes 16-31
  int rowBase = blockIdx.y * 64 + (wave >> 1) * 32;
  int colBase = blockIdx.x * 64 + (wave & 1) * 32;

  const float* arow0 = Z + (size_t)(rowBase + l16) * D + 2 * hi;
  const float* arow1 = arow0 + (size_t)16 * D;
  const float* brow0 = P + (size_t)(colBase + l16) * D + 2 * hi;
  const float* brow1 = brow0 + (size_t)16 * D;

  v8f acc00 = {0.f, 0.f, 0.f, 0.f, 0.f, 0.f, 0.f, 0.f};
  v8f acc01 = acc00, acc10 = acc00, acc11 = acc00;

  #pragma unroll 4
  for (int k = 0; k < D; k += 4) {
    v2f a0 = *(const v2f*)(arow0 + k);
    v2f a1 = *(const v2f*)(arow1 + k);
    v2f b0 = *(const v2f*)(brow0 + k);
    v2f b1 = *(const v2f*)(brow1 + k);
    // 4 WMMAs reuse 2 A-tiles x 2 B-tiles (8 FLOP per loaded byte).
    acc00 = __builtin_amdgcn_wmma_f32_16x16x4_f32(false, a0, false, b0,
                                                  (short)0, acc00, false, false);
    acc01 = __builtin_amdgcn_wmma_f32_16x16x4_f32(false, a0, false, b1,
                                                  (short)0, acc01, false, false);
    acc10 = __builtin_amdgcn_wmma_f32_16x16x4_f32(false, a1, false, b0,
                                                  (short)0, acc10, false, false);
    acc11 = __builtin_amdgcn_wmma_f32_16x16x4_f32(false, a1, false, b1,
                                                  (short)0, acc11, false, false);
  }

  // Epilogue: hyperbolic distance from dot products + cached norms.
  float c = current_c(raw_c[0]);
  float sc = sqrtf(c);
  float inv_sc = 1.0f / sc;  // dist = (2/sc) * 0.5 * log((1+a)/(1-a))
  float y2t[2];
  y2t[0] = Y2[colBase + l16];
  y2t[1] = Y2[colBase + 16 + l16];

  #pragma unroll
  for (int tr = 0; tr < 2; ++tr) {
    const float* x2p = X2 + rowBase + tr * 16 + (hi << 3);  // 8 consecutive
    #pragma unroll
    for (int tc = 0; tc < 2; ++tc) {
      const v8f* accp = (tr == 0) ? ((tc == 0) ? &acc00 : &acc01)
                                  : ((tc == 0) ? &acc10 : &acc11);
      float y2 = y2t[tc];
      int Nc = colBase + tc * 16 + l16;
      #pragma unroll
      for (int r = 0; r < 8; ++r) {
        float xy = (*accp)[r];
        float x2 = x2p[r];
        float A = 1.0f - 2.0f * c * xy + c * y2;   // coeff on (-x)
        float Bv = 1.0f - c * x2;                  // coeff on y
        float num2 = A * A * x2 + Bv * Bv * y2 - 2.0f * A * Bv * xy;
        float den = 1.0f - 2.0f * c * xy + c * c * x2 * y2;
        den = fmaxf(den, 1e-15f);
        float dn = sqrtf(fmaxf(num2, 0.0f)) / den;
        float arg = fminf(fmaxf(sc * dn, 0.0f), 1.0f - 1e-7f);
        float dist = inv_sc * logf((1.0f + arg) / (1.0f - arg));
        int M = rowBase + tr * 16 + r + (hi << 3);  // C/D layout: halves +8
        out[(size_t)M * Cn + Nc] = -dist;
      }
    }
  }
}

extern "C" void kernel_launch(void* const* d_in, const int* in_sizes, int n_in,
                              void* d_out, int out_size, void* d_ws, size_t ws_size,
                              hipStream_t stream) {
  const int D = 256;
  const float* feats = (const float*)d_in[0];
  const float* proto = (const float*)d_in[1];
  const float* raw_c = (const float*)d_in[2];
  float* out = (float*)d_out;

  int B = in_sizes[0] / D;  // 2048
  int C = in_sizes[1] / D;  // 512

  // Workspace layout (f32): Z[B*D] | P[C*D] | X2[B] | Y2[C]  (~2.63 MB)
  float* ws = (float*)d_ws;
  float* Z = ws;
  float* P = Z + (size_t)B * D;
  float* X2 = P + (size_t)C * D;
  float* Y2 = X2 + B;

  // Kernel 1: one wave per row, 8 waves per block.
  int totalWaves = B + C;
  int blocks1 = (totalWaves + 7) / 8;
  expmap0_prep<<<blocks1, 256, 0, stream>>>(feats, proto, raw_c, Z, P, X2, Y2, B, C);

  // Kernel 2: 32x32 tile per wave; 4 waves/block -> 64x64 macro-tile.
  dim3 block(32, 4, 1);
  dim3 grid(C / 64, B / 64, 1);  // (8, 32) = 256 workgroups
  hyper_gemm_dist<<<grid, block, 0, stream>>>(Z, P, X2, Y2, raw_c, out, C);
}